// GaussianRasterizer_78967268704646
// MI455X (gfx1250) — compile-verified
//
#include <hip/hip_runtime.h>
#include <hip/hip_bf16.h>
#include <math.h>

#define HH 192
#define WW 192
#define HWPIX (HH * WW)
#define NG 1024
#define TANFOVX 0.5f
#define TANFOVY 0.5f
#define NEARP 0.2f
#define LOWPASS 0.3f

typedef float v2f __attribute__((ext_vector_type(2)));
typedef float v8f __attribute__((ext_vector_type(8)));

// ---------------------------------------------------------------------------
// Kernel 1: per-gaussian projection -> conic, radii, power-poly coefficients
// power(x,y) = c0 + cx*x + cy*y + cxy*x*y + cxx*x^2 + cyy*y^2
// ---------------------------------------------------------------------------
__global__ void gs_proj_kernel(const float* __restrict__ means3D,
                               const float* __restrict__ opac,
                               const float* __restrict__ cols,
                               const float* __restrict__ scales,
                               const float* __restrict__ rots,
                               const float* __restrict__ vm,
                               float* __restrict__ keys,
                               float4* __restrict__ uA,
                               float4* __restrict__ uB,
                               float4* __restrict__ uM,
                               float* __restrict__ uZ,
                               int* __restrict__ radii_out) {
  int i = blockIdx.x * blockDim.x + threadIdx.x;
  if (i >= NG) return;

  const float fx = WW / (2.0f * TANFOVX);
  const float fy = HH / (2.0f * TANFOVY);
  const float limx = 1.3f * TANFOVX;
  const float limy = 1.3f * TANFOVY;

  float m0 = means3D[3 * i + 0], m1 = means3D[3 * i + 1], m2 = means3D[3 * i + 2];
  float tx = vm[0] * m0 + vm[1] * m1 + vm[2] * m2 + vm[3];
  float ty = vm[4] * m0 + vm[5] * m1 + vm[6] * m2 + vm[7];
  float tz = vm[8] * m0 + vm[9] * m1 + vm[10] * m2 + vm[11];

  // quaternion -> rotation
  float qr = rots[4 * i + 0], qx = rots[4 * i + 1];
  float qy = rots[4 * i + 2], qz = rots[4 * i + 3];
  float qn = rsqrtf(qr * qr + qx * qx + qy * qy + qz * qz);
  qr *= qn; qx *= qn; qy *= qn; qz *= qn;
  float R00 = 1.f - 2.f * (qy * qy + qz * qz), R01 = 2.f * (qx * qy - qr * qz), R02 = 2.f * (qx * qz + qr * qy);
  float R10 = 2.f * (qx * qy + qr * qz), R11 = 1.f - 2.f * (qx * qx + qz * qz), R12 = 2.f * (qy * qz - qr * qx);
  float R20 = 2.f * (qx * qz - qr * qy), R21 = 2.f * (qy * qz + qr * qx), R22 = 1.f - 2.f * (qx * qx + qy * qy);

  float s0 = scales[3 * i + 0], s1 = scales[3 * i + 1], s2 = scales[3 * i + 2];
  float M00 = R00 * s0, M01 = R01 * s1, M02 = R02 * s2;
  float M10 = R10 * s0, M11 = R11 * s1, M12 = R12 * s2;
  float M20 = R20 * s0, M21 = R21 * s1, M22 = R22 * s2;

  // Sigma = M * M^T
  float S00 = M00 * M00 + M01 * M01 + M02 * M02;
  float S01 = M00 * M10 + M01 * M11 + M02 * M12;
  float S02 = M00 * M20 + M01 * M21 + M02 * M22;
  float S11 = M10 * M10 + M11 * M11 + M12 * M12;
  float S12 = M10 * M20 + M11 * M21 + M12 * M22;
  float S22 = M20 * M20 + M21 * M21 + M22 * M22;

  float invz = 1.0f / tz;
  float txtz = fminf(fmaxf(tx * invz, -limx), limx);
  float tytz = fminf(fmaxf(ty * invz, -limy), limy);
  float J00 = fx * invz, J02 = -fx * txtz * invz;
  float J11 = fy * invz, J12 = -fy * tytz * invz;

  // T = J @ V[:3,:3]
  float T00 = J00 * vm[0] + J02 * vm[8];
  float T01 = J00 * vm[1] + J02 * vm[9];
  float T02 = J00 * vm[2] + J02 * vm[10];
  float T10 = J11 * vm[4] + J12 * vm[8];
  float T11 = J11 * vm[5] + J12 * vm[9];
  float T12 = J11 * vm[6] + J12 * vm[10];

  // cov2D = T Sigma T^T
  float w00 = T00 * S00 + T01 * S01 + T02 * S02;
  float w01 = T00 * S01 + T01 * S11 + T02 * S12;
  float w02 = T00 * S02 + T01 * S12 + T02 * S22;
  float x10 = T10 * S00 + T11 * S01 + T12 * S02;
  float x11 = T10 * S01 + T11 * S11 + T12 * S12;
  float x12 = T10 * S02 + T11 * S12 + T12 * S22;
  float c00 = w00 * T00 + w01 * T01 + w02 * T02;
  float c01 = w00 * T10 + w01 * T11 + w02 * T12;
  float c11 = x10 * T10 + x11 * T11 + x12 * T12;

  float a = c00 + LOWPASS;
  float b = c01;
  float c = c11 + LOWPASS;
  float det = a * c - b * b;
  bool valid = (tz > NEARP) && (det > 0.0f);
  float det_safe = (det == 0.0f) ? 1.0f : det;
  float ca = c / det_safe, cb = -b / det_safe, cc = a / det_safe;

  float mid = 0.5f * (a + c);
  float lam1 = mid + sqrtf(fmaxf(0.1f, mid * mid - det));
  radii_out[i] = valid ? (int)ceilf(3.0f * sqrtf(lam1)) : 0;

  float px = fx * tx * invz + (WW - 1) * 0.5f;
  float py = fy * ty * invz + (HH - 1) * 0.5f;

  float c0v = -0.5f * (ca * px * px + cc * py * py) - cb * px * py;
  float cxv = ca * px + cb * py;
  float cyv = cc * py + cb * px;
  float cxyv = -cb;
  float cxxv = -0.5f * ca;
  float cyyv = -0.5f * cc;
  if (!valid) { c0v = cxv = cyv = cxyv = cxxv = cyyv = 0.0f; }

  uA[i] = make_float4(c0v, cxv, cyv, cxyv);
  uB[i] = make_float4(cxxv, cyyv, 0.0f, 0.0f);
  uM[i] = make_float4(cols[3 * i + 0], cols[3 * i + 1], cols[3 * i + 2],
                      valid ? opac[i] : 0.0f);
  uZ[i] = tz;
  keys[i] = valid ? tz : __int_as_float(0x7f800000);  // +inf for invalid
}

// ---------------------------------------------------------------------------
// Kernel 2: stable rank sort by depth key (N=1024 -> O(N^2) is 1M cmps)
// ---------------------------------------------------------------------------
__global__ void gs_rank_kernel(const float* __restrict__ keys,
                               const float4* __restrict__ uA,
                               const float4* __restrict__ uB,
                               const float4* __restrict__ uM,
                               const float* __restrict__ uZ,
                               float4* __restrict__ sA,
                               float4* __restrict__ sB,
                               float4* __restrict__ sM,
                               float* __restrict__ sZ) {
  int i = blockIdx.x * blockDim.x + threadIdx.x;
  if (i >= NG) return;
  float ki = keys[i];
  int r = 0;
  for (int j = 0; j < NG; ++j) {
    float kj = keys[j];
    r += (kj < ki) || ((kj == ki) && (j < i));
  }
  sA[r] = uA[i];
  sB[r] = uB[i];
  sM[r] = uM[i];
  sZ[r] = uZ[i];
}

// ---------------------------------------------------------------------------
// Kernel 3: tiled rasterizer.
// Block = 32x16 threads = 16 waves; each wave owns one 16-pixel row of a
// 16x16 tile. All sorted gaussian data (52 KB) is staged into LDS once per
// block; the steady-state loop is pure LDS + WMMA + VALU:
//   per 16-gaussian chunk: 2x ds_load_b128 (A frags) ->
//   2x v_wmma_f32_16x16x4_f32 -> branchless 8-gaussian blend (v_exp_f32)
//   -> one shfl_xor(16) stitch of the two lane halves.
// ---------------------------------------------------------------------------
__global__ __launch_bounds__(512) void gs_raster_kernel(
    const float4* __restrict__ sA, const float4* __restrict__ sB,
    const float4* __restrict__ sM, const float* __restrict__ sZ,
    const float* __restrict__ bg, float* __restrict__ out) {
  __shared__ float4 lA[NG];   // 16 KB  (c0, cx, cy, cxy)
  __shared__ float4 lB[NG];   // 16 KB  (cxx, cyy, 0, 0)
  __shared__ float4 lM[NG];   // 16 KB  (r, g, b, opacity)
  __shared__ float lZ[NG];    //  4 KB  (depth)

  const int lane = threadIdx.x;  // 0..31
  const int row = threadIdx.y;   // 0..15
  const int tid = row * 32 + lane;

  // Stage the whole sorted gaussian table into LDS (once per block).
  for (int g = tid; g < NG; g += 512) {
    lA[g] = sA[g];
    lB[g] = sB[g];
    lM[g] = sM[g];
    lZ[g] = sZ[g];
  }
  __syncthreads();

  const int half = lane >> 4;    // 0: gaussians M0..7 | 1: M8..15
  const int col = lane & 15;     // pixel column within tile (N index)

  const float xf = (float)(blockIdx.x * 16 + col);
  const float yf = (float)(blockIdx.y * 16 + row);

  // B fragments (K x N = 4x16, loop invariant):
  //   WMMA1 features: K0=1, K1=x, K2=y, K3=x*y
  //   WMMA2 features: K0=x^2, K1=y^2, K2=0, K3=0
  v2f b1, b2;
  if (half == 0) {
    b1[0] = 1.0f;    b1[1] = xf;
    b2[0] = xf * xf; b2[1] = yf * yf;
  } else {
    b1[0] = yf;      b1[1] = xf * yf;
    b2[0] = 0.0f;    b2[1] = 0.0f;
  }

  float T_state = 1.0f;
  float cr = 0.0f, cg = 0.0f, cbl = 0.0f, dd = 0.0f;

  for (int chunk = 0; chunk < NG / 16; ++chunk) {
    const int gbase = chunk * 16;

    // A fragments (M x K = 16x4): lane half selects the K pair.
    float4 a4 = lA[gbase + col];
    float4 q4 = lB[gbase + col];
    v2f a1, a2;
    a1[0] = half ? a4.z : a4.x;  // K0: c0  | K2: cy
    a1[1] = half ? a4.w : a4.y;  // K1: cx  | K3: cxy
    a2[0] = half ? 0.0f : q4.x;  // K0: cxx | K2: 0
    a2[1] = half ? 0.0f : q4.y;  // K1: cyy | K3: 0

    v8f acc = {};
    acc = __builtin_amdgcn_wmma_f32_16x16x4_f32(false, a2, false, b2,
                                                (short)0, acc, false, false);
    acc = __builtin_amdgcn_wmma_f32_16x16x4_f32(false, a1, false, b1,
                                                (short)0, acc, false, false);

    // Branchless blend of this lane's 8 gaussians (gbase + half*8 + k).
    float Tl = 1.0f, lr = 0.0f, lg = 0.0f, lb = 0.0f, ld = 0.0f;
#pragma unroll
    for (int k = 0; k < 8; ++k) {
      const int g = gbase + half * 8 + k;
      float4 m = lM[g];
      float tz = lZ[g];
      float pw = acc[k];
      float e = __expf(fminf(pw, 0.0f));           // unconditional v_exp_f32
      float al = fminf(0.99f, m.w * e);
      al = ((pw <= 0.0f) && (al >= (1.0f / 255.0f))) ? al : 0.0f;
      float w = al * Tl;
      lr += w * m.x; lg += w * m.y; lb += w * m.z; ld += w * tz;
      Tl *= (1.0f - al);
    }

    // Stitch lane halves: lo = gaussians 0..7 (front), hi = 8..15.
    float To = __shfl_xor(Tl, 16, 32);
    float ro = __shfl_xor(lr, 16, 32);
    float go = __shfl_xor(lg, 16, 32);
    float bo = __shfl_xor(lb, 16, 32);
    float zo = __shfl_xor(ld, 16, 32);

    float Tlo = half ? To : Tl, Thi = half ? Tl : To;
    float rlo = half ? ro : lr, rhi = half ? lr : ro;
    float glo = half ? go : lg, ghi = half ? lg : go;
    float blo = half ? bo : lb, bhi = half ? lb : bo;
    float zlo = half ? zo : ld, zhi = half ? ld : zo;

    cr  += T_state * (rlo + Tlo * rhi);
    cg  += T_state * (glo + Tlo * ghi);
    cbl += T_state * (blo + Tlo * bhi);
    dd  += T_state * (zlo + Tlo * zhi);
    T_state *= Tlo * Thi;
  }

  if (half == 0) {
    const int px = blockIdx.x * 16 + col;
    const int py = blockIdx.y * 16 + row;
    const int hw = py * WW + px;
    out[0 * HWPIX + hw] = cr + bg[0] * T_state;
    out[1 * HWPIX + hw] = cg + bg[1] * T_state;
    out[2 * HWPIX + hw] = cbl + bg[2] * T_state;
    out[3 * HWPIX + NG + hw] = dd;  // depth after radii slot
  }
}

// ---------------------------------------------------------------------------
extern "C" void kernel_launch(void* const* d_in, const int* in_sizes, int n_in,
                              void* d_out, int out_size, void* d_ws,
                              size_t ws_size, hipStream_t stream) {
  const float* means3D = (const float*)d_in[0];
  // d_in[1] = means2D (unused by reference)
  const float* opac = (const float*)d_in[2];
  const float* cols = (const float*)d_in[3];
  const float* scales = (const float*)d_in[4];
  const float* rots = (const float*)d_in[5];
  const float* vm = (const float*)d_in[6];
  const float* bg = (const float*)d_in[7];

  float* ws = (float*)d_ws;
  float* keys = ws;                        // N
  float4* uA = (float4*)(ws + 1 * NG);     // 4N
  float4* uB = (float4*)(ws + 5 * NG);     // 4N
  float4* uM = (float4*)(ws + 9 * NG);     // 4N
  float* uZ = ws + 13 * NG;                // N
  float4* sA = (float4*)(ws + 14 * NG);    // 4N
  float4* sB = (float4*)(ws + 18 * NG);    // 4N
  float4* sM = (float4*)(ws + 22 * NG);    // 4N
  float* sZ = ws + 26 * NG;                // N   (27N floats ~ 108 KB)

  float* out = (float*)d_out;              // [color 3*H*W][radii N][depth H*W]
  int* radii = (int*)(out + 3 * HWPIX);

  gs_proj_kernel<<<NG / 256, 256, 0, stream>>>(means3D, opac, cols, scales,
                                               rots, vm, keys, uA, uB, uM, uZ,
                                               radii);
  gs_rank_kernel<<<NG / 256, 256, 0, stream>>>(keys, uA, uB, uM, uZ, sA, sB,
                                               sM, sZ);
  dim3 grid(WW / 16, HH / 16);
  dim3 block(32, 16);
  gs_raster_kernel<<<grid, block, 0, stream>>>(sA, sB, sM, sZ, bg, out);
}